// LGCN_9156870275400
// MI455X (gfx1250) — compile-verified
//
#include <hip/hip_runtime.h>

#define N_NODES 100000
#define N_EDGES 3200000
#define DIM     128
#define HOPS    8
#define OUTD    128
#define WCOLS   (DIM * (HOPS + 1))   // 1152

typedef __attribute__((ext_vector_type(2))) float v2f;
typedef __attribute__((ext_vector_type(8))) float v8f;

// ---------------- degree / norm ----------------

__global__ __launch_bounds__(256) void degree_kernel(const int* __restrict__ col,
                                                     float* __restrict__ deg, int e) {
    int i = blockIdx.x * blockDim.x + threadIdx.x;
    if (i < e) atomicAdd(&deg[col[i]], 1.0f);
}

__global__ __launch_bounds__(256) void dinv_kernel(float* __restrict__ deg, int n) {
    int i = blockIdx.x * blockDim.x + threadIdx.x;
    if (i < n) {
        float d = deg[i];
        deg[i] = (d > 0.0f) ? rsqrtf(fmaxf(d, 1.0f)) : 0.0f;   // in-place: deg -> dinv
    }
}

__global__ __launch_bounds__(256) void edge_norm_kernel(const int* __restrict__ row,
                                                        const int* __restrict__ col,
                                                        const float* __restrict__ dinv,
                                                        float* __restrict__ norm, int e) {
    int i = blockIdx.x * blockDim.x + threadIdx.x;
    if (i < e) norm[i] = dinv[row[i]] * dinv[col[i]];
}

// ---------------- propagation: one wave32 per edge, float4 per lane ----------------

__global__ __launch_bounds__(256) void prop_kernel(const float* __restrict__ x,
                                                   const int* __restrict__ row,
                                                   const int* __restrict__ col,
                                                   const float* __restrict__ norm,
                                                   float* __restrict__ xn, int e) {
    int wave = (blockIdx.x * blockDim.x + threadIdx.x) >> 5;
    int lane = threadIdx.x & 31;
    if (wave >= e) return;
    int r = row[wave];
    int c = col[wave];
    float w = norm[wave];
    const float4* src = (const float4*)(x + (size_t)r * DIM);   // 512B coalesced row read
    float4 v = src[lane];                                       // lane handles 4 features
    float* dst = xn + (size_t)c * DIM + lane * 4;
    atomicAdd(dst + 0, v.x * w);
    atomicAdd(dst + 1, v.y * w);
    atomicAdd(dst + 2, v.z * w);
    atomicAdd(dst + 3, v.w * w);
}

// ---------------- output init with bias ----------------

__global__ __launch_bounds__(256) void bias_init_kernel(const float* __restrict__ bias,
                                                        float* __restrict__ out, int n) {
    int i = blockIdx.x * blockDim.x + threadIdx.x;
    if (i < n) out[i] = bias[i & (OUTD - 1)];
}

// ---------------- fused per-hop GEMM accumulate: out += x @ Wk^T ----------------
// One wave32 computes one 16x16 output tile with V_WMMA_F32_16X16X4_F32,
// sweeping K=128 in steps of 4 (32 WMMA ops). fp32 preserves reference precision.

__global__ __launch_bounds__(32) void gemm_acc_kernel(const float* __restrict__ x,
                                                      const float* __restrict__ w,
                                                      float* __restrict__ out, int koff) {
    const int lane = threadIdx.x;     // 0..31, EXEC all-ones (no divergence)
    const int half = lane >> 4;       // 0: lanes 0-15, 1: lanes 16-31
    const int l16  = lane & 15;
    const size_t m0 = (size_t)blockIdx.x * 16;
    const int    n0 = blockIdx.y * 16;

    // C/D fragment layout (ISA 7.12.2): VGPR j, lanes 0-15 -> M=j, lanes 16-31 -> M=j+8
    const size_t m_base = m0 + (size_t)half * 8;
    v8f c;
#pragma unroll
    for (int j = 0; j < 8; ++j)
        c[j] = out[(m_base + j) * OUTD + n0 + l16];

    // A fragment: lane l16 holds row M = m0+l16; lane-half selects K pair {0,1}/{2,3}
    const float* xrow = x + (m0 + (size_t)l16) * DIM;
    // B fragment: lane l16 holds col N = n0+l16 of Wk^T, i.e. weight row n0+l16
    const float* wrow = w + (size_t)(n0 + l16) * WCOLS + koff;

#pragma unroll
    for (int kb = 0; kb < DIM; kb += 4) {
        v2f a = *(const v2f*)(xrow + kb + 2 * half);
        v2f b = *(const v2f*)(wrow + kb + 2 * half);
        c = __builtin_amdgcn_wmma_f32_16x16x4_f32(
                /*neg_a=*/false, a, /*neg_b=*/false, b,
                /*c_mod=*/(short)0, c, /*reuse_a=*/false, /*reuse_b=*/false);
    }

#pragma unroll
    for (int j = 0; j < 8; ++j)
        out[(m_base + j) * OUTD + n0 + l16] = c[j];
}

// ---------------- driver ----------------

extern "C" void kernel_launch(void* const* d_in, const int* in_sizes, int n_in,
                              void* d_out, int out_size, void* d_ws, size_t ws_size,
                              hipStream_t stream) {
    const float* feature    = (const float*)d_in[0];             // [N, D]
    const int*   edge_index = (const int*)d_in[1];               // [2, E]
    const float* weight     = (const float*)d_in[2];             // [OUT, D*(K+1)]
    const float* bias       = (const float*)d_in[3];             // [OUT]
    const int* row = edge_index;
    const int* col = edge_index + N_EDGES;
    float* out = (float*)d_out;

    // workspace layout (floats): deg/dinv | norm | xa | xb   (~116 MB)
    float* deg  = (float*)d_ws;
    float* norm = deg + N_NODES;
    float* xa   = norm + N_EDGES;
    float* xb   = xa + (size_t)N_NODES * DIM;

    // degrees -> dinv -> per-edge norm
    hipMemsetAsync(deg, 0, (size_t)N_NODES * sizeof(float), stream);
    degree_kernel<<<(N_EDGES + 255) / 256, 256, 0, stream>>>(col, deg, N_EDGES);
    dinv_kernel<<<(N_NODES + 255) / 256, 256, 0, stream>>>(deg, N_NODES);
    edge_norm_kernel<<<(N_EDGES + 255) / 256, 256, 0, stream>>>(row, col, deg, norm, N_EDGES);

    // out = bias (broadcast), then accumulate hop-0 GEMM
    bias_init_kernel<<<(N_NODES * OUTD + 255) / 256, 256, 0, stream>>>(bias, out, N_NODES * OUTD);

    dim3 ggrid(N_NODES / 16, OUTD / 16);   // 6250 x 8 tiles, one wave each
    gemm_acc_kernel<<<ggrid, 32, 0, stream>>>(feature, weight, out, 0);

    // K hops, each immediately folded into out (H never materialized)
    const float* xcur = feature;
    float* bufs[2] = { xa, xb };
    for (int k = 1; k <= HOPS; ++k) {
        float* xnext = bufs[(k - 1) & 1];
        hipMemsetAsync(xnext, 0, (size_t)N_NODES * DIM * sizeof(float), stream);
        prop_kernel<<<(N_EDGES + 7) / 8, 256, 0, stream>>>(xcur, row, col, norm, xnext, N_EDGES);
        gemm_acc_kernel<<<ggrid, 32, 0, stream>>>(xnext, weight, out, k * DIM);
        xcur = xnext;
    }
}